// RockFaciesClassifier_11914239279182
// MI455X (gfx1250) — compile-verified
//
#include <hip/hip_runtime.h>

typedef __attribute__((ext_vector_type(2))) float v2f;
typedef __attribute__((ext_vector_type(8))) float v8f;

#define WG 256

// ---------------- utility kernels ----------------

__global__ void zero_f32(float* __restrict__ p, size_t n) {
  size_t i = (size_t)blockIdx.x * blockDim.x + threadIdx.x;
  if (i < n) p[i] = 0.0f;
}

// deg[dst] += 1 per edge (self-loop added in deg_to_dinv)
__global__ void deg_count(const long long* __restrict__ dst, float* __restrict__ deg, int E) {
  int e = blockIdx.x * blockDim.x + threadIdx.x;
  if (e < E) atomicAdd(&deg[(size_t)dst[e]], 1.0f);
}

__global__ void deg_to_dinv(float* __restrict__ deg, int n) {
  int i = blockIdx.x * blockDim.x + threadIdx.x;
  if (i < n) deg[i] = rsqrtf(deg[i] + 1.0f);  // +1 self-loop; always > 0
}

// ---------------- GEMM 1: H1[n,128] = X[n,64] @ W1[64,128] (fp32 WMMA) ----------------
// grid.x = ceil(n/16). 256 threads = 8 waves; wave w owns N-tile w (16 cols of 128).
__global__ void gemm1_wmma(const float* __restrict__ X, const float* __restrict__ W1,
                           float* __restrict__ H1, int n) {
  __shared__ float sW[64 * 128];  // whole W1: 32 KB
  __shared__ float sX[16 * 64];   // one M-tile of X: 4 KB
  const int tid = threadIdx.x;
  const int m0 = blockIdx.x * 16;

  for (int i = tid; i < 64 * 128; i += WG) sW[i] = W1[i];
  for (int i = tid; i < 16 * 64; i += WG) {
    int r = i >> 6, c = i & 63;
    int row = m0 + r; if (row >= n) row = n - 1;   // clamp (keeps EXEC full)
    sX[i] = X[(size_t)row * 64 + c];
  }
  __syncthreads();

  const int wave = tid >> 5;
  const int lane = tid & 31;
  const int n0   = wave * 16;
  const int mrow = lane & 15;
  const int koff = (lane >> 4) * 2;   // lanes 0-15 -> K+0/1, lanes 16-31 -> K+2/3

  v8f c = {};
  #pragma unroll
  for (int k = 0; k < 64; k += 4) {
    v2f a, b;
    a.x = sX[mrow * 64 + k + koff];
    a.y = sX[mrow * 64 + k + koff + 1];
    b.x = sW[(k + koff) * 128 + n0 + mrow];
    b.y = sW[(k + koff + 1) * 128 + n0 + mrow];
    c = __builtin_amdgcn_wmma_f32_16x16x4_f32(false, a, false, b, (short)0, c, false, false);
  }

  const int mo  = m0 + (lane >> 4) * 8;
  const int col = n0 + (lane & 15);
  #pragma unroll
  for (int r = 0; r < 8; ++r) {
    int row = mo + r;
    if (row < n) H1[(size_t)row * 128 + col] = c[r];
  }
}

// ---------------- edge scatter, layer 1 (128 feat): one wave per edge ----------------
__global__ void scatter128(const long long* __restrict__ src, const long long* __restrict__ dst,
                           const float* __restrict__ dinv, const float* __restrict__ H1,
                           float* __restrict__ A1, int E) {
  int gid = blockIdx.x * blockDim.x + threadIdx.x;
  int e = gid >> 5;
  if (e >= E) return;
  int lane = gid & 31;
  long long s = src[e], d = dst[e];
  float w = dinv[s] * dinv[d];
  float4 v = ((const float4*)(H1 + (size_t)s * 128))[lane];
  float* out = A1 + (size_t)d * 128 + lane * 4;
  atomicAdd(out + 0, v.x * w);
  atomicAdd(out + 1, v.y * w);
  atomicAdd(out + 2, v.z * w);
  atomicAdd(out + 3, v.w * w);
}

// A1 = relu(A1 + dinv^2 * H1 + b1)   (self-loop + bias + activation)
__global__ void finalize1(const float* __restrict__ H1, const float* __restrict__ dinv,
                          const float* __restrict__ b1, float* __restrict__ A1, size_t total) {
  size_t i = (size_t)blockIdx.x * blockDim.x + threadIdx.x;
  if (i >= total) return;
  int node = (int)(i >> 7), f = (int)(i & 127);
  float di = dinv[node];
  float v = A1[i] + di * di * H1[i] + b1[f];
  A1[i] = v > 0.0f ? v : 0.0f;
}

// ---------------- GEMM 2: H2[n,16] = A1[n,128] @ W2[128,16] (fp32 WMMA) ----------------
// 8 waves per block, each wave owns one 16-row M-tile; K loop = 32 WMMA.
__global__ void gemm2_wmma(const float* __restrict__ A1, const float* __restrict__ W2,
                           float* __restrict__ H2, int n) {
  __shared__ float sW[128 * 16];  // 8 KB
  const int tid = threadIdx.x;
  for (int i = tid; i < 128 * 16; i += WG) sW[i] = W2[i];
  __syncthreads();

  const int wave = tid >> 5, lane = tid & 31;
  const int ntiles = (n + 15) >> 4;
  const int mtile = blockIdx.x * 8 + wave;
  if (mtile >= ntiles) return;
  const int m0 = mtile * 16;
  const int mrow = lane & 15;
  const int koff = (lane >> 4) * 2;
  int arow = m0 + mrow; if (arow >= n) arow = n - 1;

  v8f c = {};
  #pragma unroll 8
  for (int k = 0; k < 128; k += 4) {
    v2f a, b;
    a.x = A1[(size_t)arow * 128 + k + koff];
    a.y = A1[(size_t)arow * 128 + k + koff + 1];
    b.x = sW[(k + koff) * 16 + mrow];
    b.y = sW[(k + koff + 1) * 16 + mrow];
    c = __builtin_amdgcn_wmma_f32_16x16x4_f32(false, a, false, b, (short)0, c, false, false);
  }

  const int mo = m0 + (lane >> 4) * 8;
  const int col = lane & 15;
  #pragma unroll
  for (int r = 0; r < 8; ++r) {
    int row = mo + r;
    if (row < n) H2[(size_t)row * 16 + col] = c[r];
  }
}

// ---------------- edge scatter, layer 2 (16 feat): 4 lanes per edge ----------------
__global__ void scatter16(const long long* __restrict__ src, const long long* __restrict__ dst,
                          const float* __restrict__ dinv, const float* __restrict__ H2,
                          float* __restrict__ A2, int E) {
  int gid = blockIdx.x * blockDim.x + threadIdx.x;
  int e = gid >> 2;
  if (e >= E) return;
  int cidx = gid & 3;
  long long s = src[e], d = dst[e];
  float w = dinv[s] * dinv[d];
  float4 v = ((const float4*)(H2 + (size_t)s * 16))[cidx];
  float* out = A2 + (size_t)d * 16 + cidx * 4;
  atomicAdd(out + 0, v.x * w);
  atomicAdd(out + 1, v.y * w);
  atomicAdd(out + 2, v.z * w);
  atomicAdd(out + 3, v.w * w);
}

__global__ void finalize2(const float* __restrict__ H2, const float* __restrict__ dinv,
                          const float* __restrict__ b2, float* __restrict__ A2, size_t total) {
  size_t i = (size_t)blockIdx.x * blockDim.x + threadIdx.x;
  if (i >= total) return;
  int node = (int)(i >> 4), f = (int)(i & 15);
  float di = dinv[node];
  float v = A2[i] + di * di * H2[i] + b2[f];
  A2[i] = v > 0.0f ? v : 0.0f;
}

// ---------------- head: out[n,10] = A2[n,16] @ Wl[16,10] + bl ----------------
__global__ void head_kernel(const float* __restrict__ A2, const float* __restrict__ Wl,
                            const float* __restrict__ bl, float* __restrict__ out, int n) {
  int node = blockIdx.x * blockDim.x + threadIdx.x;
  if (node >= n) return;
  float a[16];
  const float4* p = (const float4*)(A2 + (size_t)node * 16);
  float4 q0 = p[0], q1 = p[1], q2 = p[2], q3 = p[3];
  a[0]=q0.x; a[1]=q0.y; a[2]=q0.z; a[3]=q0.w;
  a[4]=q1.x; a[5]=q1.y; a[6]=q1.z; a[7]=q1.w;
  a[8]=q2.x; a[9]=q2.y; a[10]=q2.z; a[11]=q2.w;
  a[12]=q3.x; a[13]=q3.y; a[14]=q3.z; a[15]=q3.w;
  #pragma unroll
  for (int j = 0; j < 10; ++j) {
    float s = bl[j];
    #pragma unroll
    for (int k = 0; k < 16; ++k) s += a[k] * Wl[k * 10 + j];
    out[(size_t)node * 10 + j] = s;
  }
}

// ---------------- launch ----------------
extern "C" void kernel_launch(void* const* d_in, const int* in_sizes, int n_in,
                              void* d_out, int out_size, void* d_ws, size_t ws_size,
                              hipStream_t stream) {
  const float*     x    = (const float*)d_in[0];
  const long long* edge = (const long long*)d_in[1];
  const float*     W1   = (const float*)d_in[2];
  const float*     b1   = (const float*)d_in[3];
  const float*     W2   = (const float*)d_in[4];
  const float*     b2   = (const float*)d_in[5];
  const float*     Wl   = (const float*)d_in[6];
  const float*     bl   = (const float*)d_in[7];
  float* out = (float*)d_out;

  const int n = in_sizes[0] / 64;       // 100000 nodes
  const int E = in_sizes[1] / 2;        // 1.6M edges
  const long long* src = edge;
  const long long* dst = edge + E;

  // workspace layout: [dinv | A1 | A2 | H1 | H2]  (first three zero-initialized)
  float* ws  = (float*)d_ws;
  float* dinv = ws;                                 // n
  float* A1   = dinv + n;                           // n*128
  float* A2   = A1 + (size_t)n * 128;               // n*16
  float* H1   = A2 + (size_t)n * 16;                // n*128
  float* H2   = H1 + (size_t)n * 128;               // n*16

  const size_t nz = (size_t)n * (1 + 128 + 16);
  zero_f32<<<(unsigned)((nz + WG - 1) / WG), WG, 0, stream>>>(ws, nz);

  deg_count<<<(E + WG - 1) / WG, WG, 0, stream>>>(dst, dinv, E);
  deg_to_dinv<<<(n + WG - 1) / WG, WG, 0, stream>>>(dinv, n);

  const int mtiles = (n + 15) / 16;
  gemm1_wmma<<<mtiles, WG, 0, stream>>>(x, W1, H1, n);

  scatter128<<<(unsigned)(((size_t)E * 32 + WG - 1) / WG), WG, 0, stream>>>(src, dst, dinv, H1, A1, E);
  finalize1<<<(unsigned)(((size_t)n * 128 + WG - 1) / WG), WG, 0, stream>>>(H1, dinv, b1, A1, (size_t)n * 128);

  gemm2_wmma<<<(mtiles + 7) / 8, WG, 0, stream>>>(A1, W2, H2, n);

  scatter16<<<(unsigned)(((size_t)E * 4 + WG - 1) / WG), WG, 0, stream>>>(src, dst, dinv, H2, A2, E);
  finalize2<<<(unsigned)(((size_t)n * 16 + WG - 1) / WG), WG, 0, stream>>>(H2, dinv, b2, A2, (size_t)n * 16);

  head_kernel<<<(n + WG - 1) / WG, WG, 0, stream>>>(A2, Wl, bl, out, n);
}